// LocalAvgPool2D_14989435863588
// MI455X (gfx1250) — compile-verified
//
#include <hip/hip_runtime.h>

typedef __attribute__((ext_vector_type(2))) float v2f;
typedef __attribute__((ext_vector_type(8))) float v8f;

static constexpr int Hh = 256;
static constexpr int Ww = 256;
static constexpr int Cc = 64;
static constexpr int CS_PITCH = 260;   // 256 + 4 pad floats: conflict-free phase-1 stores

__device__ __forceinline__ int iclamp(int v, int lo, int hi) {
    return v < lo ? lo : (v > hi ? hi : v);
}

extern "C" __global__ __launch_bounds__(256)
void local_avg_pool_cdna5(const float* __restrict__ in, float* __restrict__ out) {
    extern __shared__ float lds[];          // 256 * 260 floats = 266,240 B dynamic LDS
    const int tid  = threadIdx.x;
    const int lane = tid & 31;
    const int wave = tid >> 5;
    const int c0   = blockIdx.x * 4;        // channel block (4 channels)
    const int n    = blockIdx.y;            // image
    const size_t planeBase = (size_t)n * Hh * Ww * Cc;

    // ---------------- Phase 1: horizontal 193-wide sliding sums ----------------
    // cs[r][j][c] = sum_{w=j..j+192} x[n,r,w,c0+c]   stored at lds[r*260 + j*4 + c]
    #pragma unroll
    for (int q = 0; q < 4; ++q) {
        const int task = q * 256 + tid;     // 1024 tasks = 256 rows x 4 channels
        const int c = task & 3;
        const int r = task >> 2;
        const float* p = in + planeBase + (size_t)r * (Ww * Cc) + (c0 + c);
        float run = 0.f;
        for (int w = 0; w < 193; ++w) run += p[(size_t)w * Cc];
        float* csr = &lds[r * CS_PITCH + c];
        csr[0] = run;
        for (int j = 1; j < 64; ++j) {
            run += p[(size_t)(j + 192) * Cc] - p[(size_t)(j - 1) * Cc];
            csr[j * 4] = run;
        }
    }
    __syncthreads();

    // ---------------- Phase 2: vertical 193-wide sums via WMMA f32 16x16x4 -----
    // gap[i][nn] = sum_r Band[i][r] * cs[r][nn],  nn = j*4+c in [0,256)
    // Band[i][r] = 1 if i <= r <= i+192 else 0  (exact 0/1 weights)
    float dreg[8][8];
    const int mrow  = lane & 15;            // M within tile (A) / N within tile (B)
    const int khalf = (lane >> 4) << 1;     // lanes 0-15 -> K{0,1}, lanes 16-31 -> K{2,3}
    #pragma unroll
    for (int t = 0; t < 8; ++t) {
        const int tileId = wave * 8 + t;    // 64 tiles: 4 M-tiles x 16 N-tiles
        const int mt = tileId & 3;
        const int nt = tileId >> 2;
        const int i0 = mt * 16;
        const int nn = nt * 16 + mrow;
        v8f acc = {0.f, 0.f, 0.f, 0.f, 0.f, 0.f, 0.f, 0.f};
        for (int k0 = i0; k0 <= i0 + 204; k0 += 4) {   // band support: r in [i0, i0+207]
            const int r = k0 + khalf;
            v2f a, b;
            a.x = ((unsigned)(r     - (i0 + mrow)) <= 192u) ? 1.f : 0.f;
            a.y = ((unsigned)(r + 1 - (i0 + mrow)) <= 192u) ? 1.f : 0.f;
            b.x = lds[(size_t)r       * CS_PITCH + nn];
            b.y = lds[(size_t)(r + 1) * CS_PITCH + nn];
            acc = __builtin_amdgcn_wmma_f32_16x16x4_f32(
                false, a, false, b, (short)0, acc, false, false);
        }
        #pragma unroll
        for (int v = 0; v < 8; ++v) dreg[t][v] = acc[v];
    }
    __syncthreads();

    // ---- re-stage gap (scaled by 1/4096, exact pow2) into LDS overlay ----
    // layout: lds[i*256 + j*4 + c], 64 KB, 16B-aligned rows for float4 reads
    const float scale = 1.f / 4096.f;
    #pragma unroll
    for (int t = 0; t < 8; ++t) {
        const int tileId = wave * 8 + t;
        const int mt = tileId & 3;
        const int nt = tileId >> 2;
        const int nn = nt * 16 + mrow;
        const int mb = mt * 16 + ((lane >> 4) << 3);   // +0 (lanes 0-15) or +8 (16-31)
        #pragma unroll
        for (int v = 0; v < 8; ++v)
            lds[(mb + v) * 256 + nn] = dreg[t][v] * scale;
    }
    __syncthreads();

    // ---------------- Phase 3: edge-replicated expansion to 256x256 ------------
    float* outp = out + planeBase + c0;
    for (int y = 0; y < 256; ++y) {
        const int iy = iclamp(y - 96, 0, 63);
        const int x  = tid;
        const int ix = iclamp(x - 96, 0, 63);
        const float4* g = (const float4*)&lds[iy * 256];
        const float4 vv = g[ix];
        *(float4*)(outp + (size_t)(y * Ww + x) * Cc) = vv;   // global_store_b128
    }
}

extern "C" void kernel_launch(void* const* d_in, const int* in_sizes, int n_in,
                              void* d_out, int out_size, void* d_ws, size_t ws_size,
                              hipStream_t stream) {
    (void)in_sizes; (void)n_in; (void)out_size; (void)d_ws; (void)ws_size;
    const float* in = (const float*)d_in[0];
    float* out = (float*)d_out;
    const dim3 grid(16, 16);                 // 16 channel-blocks x 16 images
    const dim3 block(256);                   // 8 waves (wave32)
    const size_t shmem = (size_t)Hh * CS_PITCH * sizeof(float);  // 266,240 B < 320 KB
    hipLaunchKernelGGL(local_avg_pool_cdna5, grid, block, shmem, stream, in, out);
}